// graph_MHA_layer_31928786879183
// MI455X (gfx1250) — compile-verified
//
#include <hip/hip_runtime.h>
#include <math.h>

// ---------------------------------------------------------------------------
// Graph multi-head attention layer (GAT-like) for MI455X / gfx1250.
//   Phase 1: Q/K/V = h @ W + b   via V_WMMA_F32_16X16X4_F32 (full f32 WMMA)
//   Phase 2: per-edge SDDMM + exp + scatter-add (wave32 per edge, L2 atomics)
//   Phase 3: out = num / den
// N=50000, HID=128, H=8 heads, D=16 -> H*D = 128 (exactly 8 16-wide tiles,
// and N is exactly 3125 16-row tiles: no edge masking, EXEC stays all-ones).
// ---------------------------------------------------------------------------

typedef float v2f __attribute__((ext_vector_type(2)));
typedef float v8f __attribute__((ext_vector_type(8)));

#define NODES 50000
#define HIDN  128
#define NHEAD 8

// ---------------------------------------------------------------------------
// Phase 1: one wave computes one 16x16 tile of Y = X @ W + B.
//   A (16x4 f32):  lane(0..15) holds row M=lane, K=k..k+1; lanes 16..31 hold
//                  the same rows at K=k+2..k+3  -> one 8B load per lane.
//   B (4x16 f32):  VGPR v holds row K=v (lanes 0-15) / K=v+2 (lanes 16-31),
//                  column = lane&15.
//   C/D (16x16):   VGPR v = row v (lanes 0-15) / row v+8 (lanes 16-31).
// ---------------------------------------------------------------------------
__global__ void qkv_gemm_wmma(const float* __restrict__ X,
                              const float* __restrict__ W,
                              const float* __restrict__ B,
                              float* __restrict__ Y)
{
    const int nTileM = NODES / 16;                               // 3125
    const int wid = blockIdx.x * (blockDim.x >> 5) + (threadIdx.x >> 5);
    if (wid >= nTileM) return;                                   // wave-uniform

    const int lane = threadIdx.x & 31;
    const int half = lane >> 4;          // 0: K even pair, 1: K odd pair
    const int lm   = lane & 15;

    const int row  = wid * 16 + lm;                  // A-matrix row (node)
    const int ncol = blockIdx.y * 16 + lm;           // B/D column

    const float* __restrict__ xrow = X + (size_t)row * HIDN;

    v8f acc = {};
#pragma unroll 4
    for (int kk = 0; kk < HIDN; kk += 4) {
        const int k2 = kk + 2 * half;
        v2f a = *(const v2f*)(xrow + k2);            // h[row][k2..k2+1]
        v2f b;
        b.x = W[(size_t)(k2 + 0) * HIDN + ncol];     // W[k2  ][ncol]
        b.y = W[(size_t)(k2 + 1) * HIDN + ncol];     // W[k2+1][ncol]
        acc = __builtin_amdgcn_wmma_f32_16x16x4_f32(
                  false, a, false, b, (short)0, acc, false, false);
    }

    const float bias = B[ncol];
#pragma unroll
    for (int v = 0; v < 8; ++v) {
        Y[(size_t)(wid * 16 + v + 8 * half) * HIDN + ncol] = acc[v] + bias;
    }
}

// ---------------------------------------------------------------------------
// Zero-fill helper (d_out / den are poisoned by the harness).
// ---------------------------------------------------------------------------
__global__ void zero_f32(float* __restrict__ p, int n)
{
    int i = blockIdx.x * blockDim.x + threadIdx.x;
    if (i < n) p[i] = 0.0f;
}

// ---------------------------------------------------------------------------
// Phase 2: one wave32 per edge. Lane l owns dims [4l, 4l+4); head = l>>2.
// score_h = <K[src,h,:], Q[dst,h,:]> / sqrt(16); w = exp(score);
// out[dst] += w * V[src] (f32 atomics, resolved in L2); den[dst,h] += w.
// Q/K/V total 76.8 MB < 192 MB L2, so the random gathers stay on-chip.
// ---------------------------------------------------------------------------
__global__ void edge_attn(const float4* __restrict__ Q4,
                          const float4* __restrict__ K4,
                          const float4* __restrict__ V4,
                          const int*    __restrict__ src,
                          const int*    __restrict__ dst,
                          float*        __restrict__ out,
                          float*        __restrict__ den,
                          int n_edges)
{
    const int e = (blockIdx.x * blockDim.x + threadIdx.x) >> 5;
    if (e >= n_edges) return;                        // wave-uniform
    const int lane = threadIdx.x & 31;

    const int s = src[e];
    const int d = dst[e];

    // coalesced 512B gathers: 32 lanes x float4 per row of 128 floats
    const float4 kv = K4[(size_t)s * 32 + lane];
    const float4 qv = Q4[(size_t)d * 32 + lane];

    float p = kv.x * qv.x + kv.y * qv.y + kv.z * qv.z + kv.w * qv.w;
    // reduce within each head's 4-lane group
    p += __shfl_xor(p, 1, 4);
    p += __shfl_xor(p, 2, 4);

    const float w = expf(p * 0.25f);                 // 1/sqrt(D) = 1/4

    const float4 vv = V4[(size_t)s * 32 + lane];
    float* o = out + (size_t)d * HIDN + lane * 4;
    atomicAdd(o + 0, w * vv.x);
    atomicAdd(o + 1, w * vv.y);
    atomicAdd(o + 2, w * vv.z);
    atomicAdd(o + 3, w * vv.w);

    if ((lane & 3) == 0)                             // one lane per head
        atomicAdd(den + (size_t)d * NHEAD + (lane >> 2), w);
}

// ---------------------------------------------------------------------------
// Phase 3: out[n,h,:] /= den[n,h]
// ---------------------------------------------------------------------------
__global__ void normalize_out(float* __restrict__ out,
                              const float* __restrict__ den)
{
    int i = blockIdx.x * blockDim.x + threadIdx.x;
    if (i < NODES * HIDN) {
        int n = i >> 7;                // / 128
        int h = (i >> 4) & 7;          // head
        out[i] /= den[n * NHEAD + h];
    }
}

// ---------------------------------------------------------------------------
extern "C" void kernel_launch(void* const* d_in, const int* in_sizes, int n_in,
                              void* d_out, int out_size, void* d_ws, size_t ws_size,
                              hipStream_t stream)
{
    const float* h  = (const float*)d_in[0];
    const float* wq = (const float*)d_in[1];
    const float* bq = (const float*)d_in[2];
    const float* wk = (const float*)d_in[3];
    const float* bk = (const float*)d_in[4];
    const float* wv = (const float*)d_in[5];
    const float* bv = (const float*)d_in[6];
    const int*   src = (const int*)d_in[7];
    const int*   dst = (const int*)d_in[8];
    const int E = in_sizes[7];

    float* out = (float*)d_out;

    // workspace: Q | K | V | den  (3*25.6MB + 1.6MB)
    char* ws = (char*)d_ws;
    const size_t matB = (size_t)NODES * HIDN * sizeof(float);
    float* Q   = (float*)(ws);
    float* K   = (float*)(ws + matB);
    float* V   = (float*)(ws + 2 * matB);
    float* den = (float*)(ws + 3 * matB);

    // --- Phase 1: QKV projections (f32 WMMA) ---
    const int nTileM = NODES / 16;                     // 3125
    dim3 gblk(128);                                    // 4 waves / block
    dim3 ggrid((nTileM + 3) / 4, HIDN / 16);           // y = column tile (8)
    qkv_gemm_wmma<<<ggrid, gblk, 0, stream>>>(h, wq, bq, Q);
    qkv_gemm_wmma<<<ggrid, gblk, 0, stream>>>(h, wk, bk, K);
    qkv_gemm_wmma<<<ggrid, gblk, 0, stream>>>(h, wv, bv, V);

    // --- zero accumulators (num in d_out, den in ws) ---
    zero_f32<<<(NODES * HIDN + 255) / 256, 256, 0, stream>>>(out, NODES * HIDN);
    zero_f32<<<(NODES * NHEAD + 255) / 256, 256, 0, stream>>>(den, NODES * NHEAD);

    // --- Phase 2: edge SDDMM + weighted scatter (1 wave per edge) ---
    {
        long long threads = (long long)E * 32;
        int blocks = (int)((threads + 255) / 256);
        edge_attn<<<blocks, 256, 0, stream>>>((const float4*)Q, (const float4*)K,
                                              (const float4*)V, src, dst,
                                              out, den, E);
    }

    // --- Phase 3: normalize ---
    normalize_out<<<(NODES * HIDN + 255) / 256, 256, 0, stream>>>(out, den);
}